// DisModelPerLevel_45878840656483
// MI455X (gfx1250) — compile-verified
//
#include <hip/hip_runtime.h>
#include <hip/hip_bf16.h>
#include <math.h>

// ---------------------------------------------------------------------------
// Problem constants (from reference: N=8, C=256, H=W=128, windows [3,66,128])
// ---------------------------------------------------------------------------
#define NB 8
#define CH 256
#define C2 512
#define HH 128
#define WW 128
#define OUT0 126              // (128-3)/1 + 1
#define P0 (OUT0*OUT0)        // 15876
#define NK 24                 // N * NUM_WINDOWS rows
#define INV_SCALE 0.0625f     // 1/sqrt(256)

// Workspace layout (fp32 elements)
enum : int {
  OFF_X1      = 0,                         // [8][512][4]
  OFF_X2      = OFF_X1 + NB*C2*4,          // [8][512]
  OFF_CENTERS = OFF_X2 + NB*C2,            // [3][8][512]
  OFF_THETA   = OFF_CENTERS + 3*NB*C2,     // [24][256]
  OFF_T2      = OFF_THETA + NK*CH,         // [3][24][512]
  OFF_PBDOT   = OFF_T2 + 3*NK*C2,          // [3][24]
  OFF_PWLOG   = OFF_PBDOT + 3*NK,          // [24][15876]
  OFF_BMAX    = OFF_PWLOG + NK*P0,         // [24][64]
  OFF_MAXV    = OFF_BMAX + NK*64,          // [24]
  OFF_SUMEXP  = OFF_MAXV + NK,             // [24]
  OFF_RES     = OFF_SUMEXP + NK,           // [3][24][512]
  OFF_H1      = OFF_RES + 3*NK*C2,         // [3][24][256]
  OFF_H2      = OFF_H1 + 3*NK*CH,          // [3][24][256]
  OFF_H3      = OFF_H2 + 3*NK*CH,          // [3][24][128]
  WS_TOTAL    = OFF_H3 + 3*NK*128
};

typedef __attribute__((ext_vector_type(2))) float v2f;
typedef __attribute__((ext_vector_type(8))) float v8f;

__device__ inline float leaky02(float v) { return v >= 0.f ? v : 0.2f * v; }

// 3x3 box mean + biased std at pp (row stride `stride`)
__device__ inline float box3_mean_std(const float* pp, int stride, float* sd) {
  float s = 0.f, s2 = 0.f;
#pragma unroll
  for (int dr = 0; dr < 3; ++dr) {
    float a = pp[dr*stride+0], b = pp[dr*stride+1], d = pp[dr*stride+2];
    s  += a + b + d;
    s2 += a*a + b*b + d*d;
  }
  float m = s * (1.f/9.f);
  float var = fmaxf(s2 * (1.f/9.f) - m*m, 0.f);
  *sd = sqrtf(var + 1e-12f);
  return m;
}

// ---------------------------------------------------------------------------
// WMMA helper: one wave32 computes a 16x16 fp32 tile, K-loop over fp32 data.
// A fed via la(m,k), B via lb(k,n). Uses V_WMMA_F32_16X16X4_F32.
// NOTE: la/lb must be UNCONDITIONAL loads (rows of the output are independent,
// so pad-row garbage only lands in pad output rows which the epilogue drops).
// This keeps the inner loop free of exec-mask diamonds.
// ---------------------------------------------------------------------------
template <typename FA, typename FB>
__device__ inline v8f wmma_f32_gemm(int K, FA la, FB lb) {
  v8f acc = {};
  const int lane = threadIdx.x & 31;
  const int half = lane >> 4;
  const int mn   = lane & 15;
#pragma unroll 8
  for (int kk = 0; kk < K; kk += 4) {
    const int ka = kk + 2*half;
    v2f a, b;
    a.x = la(mn, ka);     a.y = la(mn, ka + 1);
    b.x = lb(ka, mn);     b.y = lb(ka + 1, mn);
    acc = __builtin_amdgcn_wmma_f32_16x16x4_f32(false, a, false, b,
                                                (short)0, acc, false, false);
  }
  return acc;
}

// ---------------------------------------------------------------------------
// K1: per (n,c) plane -> windows 1/2 maps (x1,x2), all three center vectors.
// Plane staged in 64KB LDS, reused as reduction scratch. No atomics.
// ---------------------------------------------------------------------------
__global__ void k1_plane_stats(const float* __restrict__ feat, float* __restrict__ ws) {
  extern __shared__ float plane[];            // 16384 floats, reused for reduce
  const int pid = blockIdx.x;                 // n*256 + c
  const int n = pid >> 8, c = pid & 255;
  const int tid = threadIdx.x;                // 256 threads
  const float* src = feat + (size_t)pid * (HH*WW);
  for (int i = tid; i < HH*WW; i += 256) plane[i] = src[i];
  __syncthreads();

  float acc[12];
#pragma unroll
  for (int j = 0; j < 12; ++j) acc[j] = 0.f;

  // [0]=full sum [1]=full sq  [2..5]=66x66 box sums  [6..9]=box sq
  for (int i = tid; i < HH*WW; i += 256) {
    float v = plane[i], v2 = v*v;
    int r = i >> 7, cc = i & 127;
    acc[0] += v; acc[1] += v2;
    bool ra = r < 66, rb = (r >= 32) & (r < 98);
    bool ca = cc < 66, cb = (cc >= 32) & (cc < 98);
    if (ra & ca) { acc[2] += v; acc[6] += v2; }
    if (ra & cb) { acc[3] += v; acc[7] += v2; }
    if (rb & ca) { acc[4] += v; acc[8] += v2; }
    if (rb & cb) { acc[5] += v; acc[9] += v2; }
  }
  // [10]=sum of window-0 means, [11]=sum of window-0 stds
  for (int o = tid; o < P0; o += 256) {
    int orow = o / OUT0, ocol = o - orow * OUT0;
    float sd;
    float m = box3_mean_std(plane + orow*WW + ocol, WW, &sd);
    acc[10] += m; acc[11] += sd;
  }
  __syncthreads();   // plane reads done; safe to reuse as scratch

  float tot[12];
  for (int j = 0; j < 12; ++j) {
    plane[tid] = acc[j]; __syncthreads();
    for (int s = 128; s > 0; s >>= 1) {
      if (tid < s) plane[tid] += plane[tid + s];
      __syncthreads();
    }
    tot[j] = plane[0]; __syncthreads();
  }

  if (tid == 0) {
    float* x1 = ws + OFF_X1;
    float* x2 = ws + OFF_X2;
    float* centers = ws + OFF_CENTERS;
    // window 2 (K=128): single output == its own center
    {
      float m  = tot[0] * (1.f/16384.f);
      float m2 = tot[1] * (1.f/16384.f);
      float sd = sqrtf(fmaxf(m2 - m*m, 0.f) + 1e-12f);
      x2[n*C2 + c] = m;        x2[n*C2 + 256 + c] = sd;
      centers[(2*NB + n)*C2 + c] = m;
      centers[(2*NB + n)*C2 + 256 + c] = sd;
    }
    // window 1 (K=66, stride 32): 2x2 outputs
    float msum = 0.f, ssum = 0.f;
    for (int p = 0; p < 4; ++p) {
      float m  = tot[2+p] * (1.f/4356.f);
      float m2 = tot[6+p] * (1.f/4356.f);
      float sd = sqrtf(fmaxf(m2 - m*m, 0.f) + 1e-12f);
      x1[(n*C2 + c)*4 + p] = m;
      x1[(n*C2 + 256 + c)*4 + p] = sd;
      msum += m; ssum += sd;
    }
    centers[(1*NB + n)*C2 + c] = msum * 0.25f;
    centers[(1*NB + n)*C2 + 256 + c] = ssum * 0.25f;
    // window 0 centers
    centers[(0*NB + n)*C2 + c] = tot[10] * (1.f/(float)P0);
    centers[(0*NB + n)*C2 + 256 + c] = tot[11] * (1.f/(float)P0);
  }
}

// ---------------------------------------------------------------------------
// K3: theta_x[r, c] = sum_d centers[kw(r), n(r), d] * theta_w[c, d] + theta_b
// grid 32 waves: 2 M-tiles x 16 N-tiles, K=512. Row r = n*3 + kw.
// ---------------------------------------------------------------------------
__global__ void k3_theta(const float* __restrict__ theta_w,
                         const float* __restrict__ theta_b,
                         float* __restrict__ ws) {
  const int mt = blockIdx.x >> 4, nt = blockIdx.x & 15;
  const float* centers = ws + OFF_CENTERS;
  auto la = [=](int m, int k) -> float {
    int r = mt*16 + m;                 // pad rows read in-workspace garbage
    int nn = r / 3, kw = r - nn*3;
    return centers[(kw*NB + nn)*C2 + k];
  };
  auto lb = [=](int k, int nn) -> float {
    return theta_w[(nt*16 + nn)*C2 + k];
  };
  v8f acc = wmma_f32_gemm(C2, la, lb);
  const int lane = threadIdx.x & 31, half = lane >> 4, col = lane & 15;
  float* th = ws + OFF_THETA;
#pragma unroll
  for (int j = 0; j < 8; ++j) {
    int row = mt*16 + j + 8*half;
    if (row < NK)
      th[row*CH + nt*16 + col] = acc[j] + theta_b[nt*16 + col];
  }
}

// ---------------------------------------------------------------------------
// K4: t2[i, r, d] = sum_c theta_x[r, c] * phi_w[i, c, d]  (the key refactor)
// grid 192 waves: 3 windows x 2 M-tiles x 32 N-tiles, K=256.
// ---------------------------------------------------------------------------
__global__ void k4_t2(const float* __restrict__ phi_w, float* __restrict__ ws) {
  const int i = blockIdx.x / 64;
  const int rem = blockIdx.x - i*64;
  const int mt = rem >> 5, nt = rem & 31;
  const float* th = ws + OFF_THETA;
  const float* pw = phi_w + i*CH*C2;
  auto la = [=](int m, int k) -> float {
    return th[(mt*16 + m)*CH + k];     // unconditional; pad rows discarded
  };
  auto lb = [=](int k, int nn) -> float {
    return pw[k*C2 + nt*16 + nn];
  };
  v8f acc = wmma_f32_gemm(CH, la, lb);
  const int lane = threadIdx.x & 31, half = lane >> 4, col = lane & 15;
  float* t2 = ws + OFF_T2;
#pragma unroll
  for (int j = 0; j < 8; ++j) {
    int row = mt*16 + j + 8*half;
    if (row < NK) t2[(i*NK + row)*C2 + nt*16 + col] = acc[j];
  }
}

// K4b: pbdot[i, r] = theta_x[r, :] . phi_b[i, :]
__global__ void k4b_pbdot(const float* __restrict__ phi_b, float* __restrict__ ws) {
  int t = threadIdx.x;
  if (t >= 3*NK) return;
  int i = t / NK, r = t - i*NK;
  const float* th = ws + OFF_THETA + r*CH;
  const float* pb = phi_b + i*CH;
  float acc = 0.f;
  for (int c = 0; c < CH; ++c) acc += th[c] * pb[c];
  ws[OFF_PBDOT + t] = acc;
}

// ---------------------------------------------------------------------------
// K5: window-0 attention logits. One WG per (n, 16x16 pixel tile); per-channel
// 18x18 halo tile staged in LDS; t2 rows for this n staged in LDS.
// logit[n,k,p] = (sum_d t2[0,n*3+k,d]*x0[d,p] + pbdot) / 16
// ---------------------------------------------------------------------------
__global__ void k5_pw_logits(const float* __restrict__ feat, float* __restrict__ ws) {
  __shared__ float ftile[18*18];
  __shared__ float t2l[3*C2];
  __shared__ float red[256];
  const int n = blockIdx.y;
  const int tile = blockIdx.x;                 // 0..63
  const int ty = tile >> 3, tx = tile & 7;
  const int tid = threadIdx.x;                 // 256 threads
  const int lr = tid >> 4, lc = tid & 15;
  const int pr = ty*16 + lr, pc = tx*16 + lc;
  const bool valid = (pr < OUT0) && (pc < OUT0);

  const float* t2g = ws + OFF_T2 + (0*NK + n*3)*C2;   // 3 contiguous rows
  for (int i = tid; i < 3*C2; i += 256) t2l[i] = t2g[i];
  __syncthreads();

  float l0 = 0.f, l1 = 0.f, l2 = 0.f;
  const int r0 = ty*16, c0 = tx*16;
  for (int ch = 0; ch < CH; ++ch) {
    const float* src = feat + ((size_t)(n*CH + ch) << 14);
    // software prefetch of next channel's tile (one cacheline per thread)
    if (ch + 1 < CH)
      __builtin_prefetch(src + (HH*WW) + r0*WW + c0 + (tid & 63) * 16, 0, 1);
    for (int i = tid; i < 324; i += 256) {
      int ir = i / 18, ic = i - ir*18;
      int gr = r0 + ir, gc = c0 + ic;
      ftile[i] = (gr < HH && gc < WW) ? src[gr*WW + gc] : 0.f;
    }
    __syncthreads();
    if (valid) {
      float sd;
      float m = box3_mean_std(ftile + lr*18 + lc, 18, &sd);
      l0 += t2l[ch]*m        + t2l[256+ch]*sd;
      l1 += t2l[512+ch]*m    + t2l[768+ch]*sd;
      l2 += t2l[1024+ch]*m   + t2l[1280+ch]*sd;
    }
    __syncthreads();
  }

  const float* pbdot = ws + OFF_PBDOT;            // window 0 slice
  float lk[3] = { (l0 + pbdot[n*3+0]) * INV_SCALE,
                  (l1 + pbdot[n*3+1]) * INV_SCALE,
                  (l2 + pbdot[n*3+2]) * INV_SCALE };
  float* pwlog = ws + OFF_PWLOG;
  float* bmax  = ws + OFF_BMAX;
  for (int k = 0; k < 3; ++k) {
    if (valid) pwlog[(size_t)(n*3 + k)*P0 + pr*OUT0 + pc] = lk[k];
    red[tid] = valid ? lk[k] : -3.4e38f;
    __syncthreads();
    for (int s = 128; s > 0; s >>= 1) {
      if (tid < s) red[tid] = fmaxf(red[tid], red[tid + s]);
      __syncthreads();
    }
    if (tid == 0) bmax[(n*3 + k)*64 + tile] = red[0];
    __syncthreads();
  }
}

// K6: per (n,k) row: global max over 64 block maxima, then sum of exp.
__global__ void k6_softmax_stats(float* __restrict__ ws) {
  __shared__ float red[256];
  const int row = blockIdx.x;                 // 0..23
  const int tid = threadIdx.x;
  const float* bmax = ws + OFF_BMAX + row*64;
  red[tid] = (tid < 64) ? bmax[tid] : -3.4e38f;
  __syncthreads();
  for (int s = 128; s > 0; s >>= 1) {
    if (tid < s) red[tid] = fmaxf(red[tid], red[tid + s]);
    __syncthreads();
  }
  float mx = red[0]; __syncthreads();
  const float* lg = ws + OFF_PWLOG + (size_t)row*P0;
  float acc = 0.f;
  for (int p = tid; p < P0; p += 256) acc += expf(lg[p] - mx);
  red[tid] = acc; __syncthreads();
  for (int s = 128; s > 0; s >>= 1) {
    if (tid < s) red[tid] += red[tid + s];
    __syncthreads();
  }
  if (tid == 0) { ws[OFF_MAXV + row] = mx; ws[OFF_SUMEXP + row] = red[0]; }
}

// ---------------------------------------------------------------------------
// K7: window-0 residual. One WG per (n,c) plane: recompute box mean/std from
// LDS, weight by softmax probs, reduce over all 15876 pixels.
// res[0, n*3+k, d] = sum_p w[k,p]*x0[d,p] - centers[k, n, d]
// ---------------------------------------------------------------------------
__global__ void k7_res0(const float* __restrict__ feat, float* __restrict__ ws) {
  extern __shared__ float plane[];            // 16384 floats, reused for reduce
  const int pid = blockIdx.x;
  const int n = pid >> 8, c = pid & 255;
  const int tid = threadIdx.x;
  const float* src = feat + (size_t)pid * (HH*WW);
  for (int i = tid; i < HH*WW; i += 256) plane[i] = src[i];
  __syncthreads();

  float mx[3], inv[3];
#pragma unroll
  for (int k = 0; k < 3; ++k) {
    mx[k]  = ws[OFF_MAXV + n*3 + k];
    inv[k] = 1.f / ws[OFF_SUMEXP + n*3 + k];
  }
  const float* lg0 = ws + OFF_PWLOG + (size_t)(n*3)*P0;
  const float* lg1 = lg0 + P0;
  const float* lg2 = lg1 + P0;

  float am[3] = {0.f,0.f,0.f}, as[3] = {0.f,0.f,0.f};
  for (int o = tid; o < P0; o += 256) {
    int orow = o / OUT0, ocol = o - orow*OUT0;
    float sd;
    float m = box3_mean_std(plane + orow*WW + ocol, WW, &sd);
    float e0 = expf(lg0[o] - mx[0]);
    float e1 = expf(lg1[o] - mx[1]);
    float e2 = expf(lg2[o] - mx[2]);
    am[0] += e0*m; as[0] += e0*sd;
    am[1] += e1*m; as[1] += e1*sd;
    am[2] += e2*m; as[2] += e2*sd;
  }
  __syncthreads();   // plane reads done; reuse as reduction scratch

  float tot[6];
  for (int j = 0; j < 6; ++j) {
    plane[tid] = (j < 3) ? am[j] : as[j-3];
    __syncthreads();
    for (int s = 128; s > 0; s >>= 1) {
      if (tid < s) plane[tid] += plane[tid + s];
      __syncthreads();
    }
    tot[j] = plane[0]; __syncthreads();
  }
  if (tid == 0) {
    float* res = ws + OFF_RES;
    const float* centers = ws + OFF_CENTERS;
    for (int k = 0; k < 3; ++k) {
      int r = n*3 + k;
      res[(0*NK + r)*C2 + c]       = tot[k]   * inv[k] - centers[(k*NB + n)*C2 + c];
      res[(0*NK + r)*C2 + 256 + c] = tot[3+k] * inv[k] - centers[(k*NB + n)*C2 + 256 + c];
    }
  }
}

// ---------------------------------------------------------------------------
// K8: windows 1 (P=4) and 2 (P=1) residuals, exact tiny attention per n.
// ---------------------------------------------------------------------------
__global__ void k8_res12(float* __restrict__ ws) {
  __shared__ float x1l[C2*4];
  __shared__ float x2l[C2];
  __shared__ float wts[12];
  const int n = blockIdx.x;
  const int tid = threadIdx.x;                // 256
  const float* x1 = ws + OFF_X1 + n*C2*4;
  const float* x2 = ws + OFF_X2 + n*C2;
  for (int i = tid; i < C2*4; i += 256) x1l[i] = x1[i];
  for (int i = tid; i < C2;   i += 256) x2l[i] = x2[i];
  __syncthreads();

  if (tid < 12) {
    int k = tid >> 2, p = tid & 3;
    const float* t2r = ws + OFF_T2 + (1*NK + n*3 + k)*C2;
    float acc = 0.f;
    for (int d = 0; d < C2; ++d) acc += t2r[d] * x1l[d*4 + p];
    wts[tid] = (acc + ws[OFF_PBDOT + 1*NK + n*3 + k]) * INV_SCALE;
  }
  __syncthreads();
  if (tid < 3) {
    float a = wts[tid*4+0], b = wts[tid*4+1], c = wts[tid*4+2], d = wts[tid*4+3];
    float m = fmaxf(fmaxf(a, b), fmaxf(c, d));
    float e0 = expf(a-m), e1 = expf(b-m), e2 = expf(c-m), e3 = expf(d-m);
    float is = 1.f / (e0+e1+e2+e3);
    wts[tid*4+0] = e0*is; wts[tid*4+1] = e1*is;
    wts[tid*4+2] = e2*is; wts[tid*4+3] = e3*is;
  }
  __syncthreads();

  float* res = ws + OFF_RES;
  const float* centers = ws + OFF_CENTERS;
  for (int idx = tid; idx < 3*C2; idx += 256) {
    int k = idx >> 9, d = idx & 511;
    int r = n*3 + k;
    float cen = centers[(k*NB + n)*C2 + d];
    float acc = wts[k*4+0]*x1l[d*4+0] + wts[k*4+1]*x1l[d*4+1]
              + wts[k*4+2]*x1l[d*4+2] + wts[k*4+3]*x1l[d*4+3];
    res[(1*NK + r)*C2 + d] = acc - cen;        // window 1
    res[(2*NK + r)*C2 + d] = x2l[d] - cen;     // window 2 (softmax over 1 == 1)
  }
}

// K9: L2-normalize each of the 72 residual rows (len 512), in place.
__global__ void k9_normalize(float* __restrict__ ws) {
  __shared__ float red[256];
  const int row = blockIdx.x;                 // 0..71
  const int tid = threadIdx.x;                // 256
  float* r = ws + OFF_RES + row*C2;
  float v0 = r[tid], v1 = r[tid + 256];
  red[tid] = v0*v0 + v1*v1;
  __syncthreads();
  for (int s = 128; s > 0; s >>= 1) {
    if (tid < s) red[tid] += red[tid + s];
    __syncthreads();
  }
  float inv = 1.f / fmaxf(sqrtf(red[0]), 1e-12f);
  r[tid] = v0 * inv; r[tid + 256] = v1 * inv;
}

// ---------------------------------------------------------------------------
// K10a/b/c: MLP layers via WMMA (M=24 rows padded to 2 tiles).
// ---------------------------------------------------------------------------
__global__ void k10a_h1(const float* __restrict__ w1, const float* __restrict__ b1,
                        float* __restrict__ ws) {
  const int i = blockIdx.x / 32;
  const int rem = blockIdx.x - i*32;
  const int mt = rem >> 4, nt = rem & 15;     // N=256 -> 16 tiles, K=512
  const float* A = ws + OFF_RES + i*NK*C2;
  const float* Bm = w1 + i*CH*C2;
  auto la = [=](int m, int k) -> float { return A[(mt*16 + m)*C2 + k]; };
  auto lb = [=](int k, int nn) -> float { return Bm[(nt*16 + nn)*C2 + k]; };
  v8f acc = wmma_f32_gemm(C2, la, lb);
  const int lane = threadIdx.x & 31, half = lane >> 4, col = lane & 15;
  float* h1 = ws + OFF_H1 + i*NK*CH;
#pragma unroll
  for (int j = 0; j < 8; ++j) {
    int row = mt*16 + j + 8*half;
    if (row < NK)
      h1[row*CH + nt*16 + col] = leaky02(acc[j] + b1[i*CH + nt*16 + col]);
  }
}

__global__ void k10b_h2(const float* __restrict__ w2, const float* __restrict__ b2,
                        float* __restrict__ ws) {
  const int i = blockIdx.x / 32;
  const int rem = blockIdx.x - i*32;
  const int mt = rem >> 4, nt = rem & 15;     // N=256, K=256
  const float* A = ws + OFF_H1 + i*NK*CH;
  const float* Bm = w2 + i*CH*CH;
  auto la = [=](int m, int k) -> float { return A[(mt*16 + m)*CH + k]; };
  auto lb = [=](int k, int nn) -> float { return Bm[(nt*16 + nn)*CH + k]; };
  v8f acc = wmma_f32_gemm(CH, la, lb);
  const int lane = threadIdx.x & 31, half = lane >> 4, col = lane & 15;
  float* h2 = ws + OFF_H2 + i*NK*CH;
#pragma unroll
  for (int j = 0; j < 8; ++j) {
    int row = mt*16 + j + 8*half;
    if (row < NK)
      h2[row*CH + nt*16 + col] = leaky02(acc[j] + b2[i*CH + nt*16 + col]);
  }
}

__global__ void k10c_h3(const float* __restrict__ w3, const float* __restrict__ b3,
                        float* __restrict__ ws) {
  const int i = blockIdx.x / 16;
  const int rem = blockIdx.x - i*16;
  const int mt = rem >> 3, nt = rem & 7;      // N=128 -> 8 tiles, K=256
  const float* A = ws + OFF_H2 + i*NK*CH;
  const float* Bm = w3 + i*128*CH;
  auto la = [=](int m, int k) -> float { return A[(mt*16 + m)*CH + k]; };
  auto lb = [=](int k, int nn) -> float { return Bm[(nt*16 + nn)*CH + k]; };
  v8f acc = wmma_f32_gemm(CH, la, lb);
  const int lane = threadIdx.x & 31, half = lane >> 4, col = lane & 15;
  float* h3 = ws + OFF_H3 + i*NK*128;
#pragma unroll
  for (int j = 0; j < 8; ++j) {
    int row = mt*16 + j + 8*half;
    if (row < NK)
      h3[row*128 + nt*16 + col] = leaky02(acc[j] + b3[i*128 + nt*16 + col]);
  }
}

// K10d: final logits (N=1 column) + mean-BCE per window, sum over windows.
__global__ void k10d_bce(const float* __restrict__ w4, const float* __restrict__ b4,
                         const int* __restrict__ label,
                         float* __restrict__ ws, float* __restrict__ out) {
  __shared__ float red[128];
  const int tid = threadIdx.x;                // 128
  float term = 0.f;
  if (tid < 3*NK) {
    int i = tid / NK, r = tid - i*NK;
    const float* h3 = ws + OFF_H3 + (i*NK + r)*128;
    const float* wv = w4 + i*128;
    float l = b4[i];
    for (int d = 0; d < 128; ++d) l += h3[d] * wv[d];
    float y = (float)label[0];
    term = (fmaxf(l, 0.f) - l*y + log1pf(expf(-fabsf(l)))) * (1.f/(float)NK);
  }
  red[tid] = term; __syncthreads();
  for (int s = 64; s > 0; s >>= 1) {
    if (tid < s) red[tid] += red[tid + s];
    __syncthreads();
  }
  if (tid == 0) out[0] = red[0];
}

// ---------------------------------------------------------------------------
extern "C" void kernel_launch(void* const* d_in, const int* in_sizes, int n_in,
                              void* d_out, int out_size, void* d_ws, size_t ws_size,
                              hipStream_t stream) {
  const float* feature = (const float*)d_in[0];
  const float* theta_w = (const float*)d_in[1];
  const float* theta_b = (const float*)d_in[2];
  const float* phi_w   = (const float*)d_in[3];
  const float* phi_b   = (const float*)d_in[4];
  const float* w1      = (const float*)d_in[5];
  const float* b1      = (const float*)d_in[6];
  const float* w2      = (const float*)d_in[7];
  const float* b2      = (const float*)d_in[8];
  const float* w3      = (const float*)d_in[9];
  const float* b3      = (const float*)d_in[10];
  const float* w4      = (const float*)d_in[11];
  const float* b4      = (const float*)d_in[12];
  const int*   label   = (const int*)d_in[13];
  float* out = (float*)d_out;
  float* ws  = (float*)d_ws;

  const size_t plane_smem = (size_t)HH * WW * sizeof(float);   // 64 KB

  k1_plane_stats<<<NB*CH, 256, plane_smem, stream>>>(feature, ws);
  k3_theta      <<<32, 32, 0, stream>>>(theta_w, theta_b, ws);
  k4_t2         <<<192, 32, 0, stream>>>(phi_w, ws);
  k4b_pbdot     <<<1, 128, 0, stream>>>(phi_b, ws);
  k5_pw_logits  <<<dim3(64, NB), 256, 0, stream>>>(feature, ws);
  k6_softmax_stats<<<NK, 256, 0, stream>>>(ws);
  k7_res0       <<<NB*CH, 256, plane_smem, stream>>>(feature, ws);
  k8_res12      <<<NB, 256, 0, stream>>>(ws);
  k9_normalize  <<<3*NK, 256, 0, stream>>>(ws);
  k10a_h1       <<<96, 32, 0, stream>>>(w1, b1, ws);
  k10b_h2       <<<96, 32, 0, stream>>>(w2, b2, ws);
  k10c_h3       <<<48, 32, 0, stream>>>(w3, b3, ws);
  k10d_bce      <<<1, 128, 0, stream>>>(w4, b4, label, ws, out);
}